// GlobalGuidedLayer_72705206387095
// MI455X (gfx1250) — compile-verified
//
#include <hip/hip_runtime.h>
#include <hip/hip_bf16.h>

// MI455X / gfx1250, wave32. FP32 WMMA path (V_WMMA_F32_16X16X4_F32):
// the op is HBM-bandwidth bound (~768MB @ 23.3TB/s ~= 33us) with only
// ~17 GFLOP, so fp32 matrix cores are the right precision/data path.

typedef __attribute__((ext_vector_type(2))) float v2f;
typedef __attribute__((ext_vector_type(8))) float v8f;

#define WMMA_F32X4(A, B, C) \
  __builtin_amdgcn_wmma_f32_16x16x4_f32(false, (A), false, (B), (short)0, (C), false, false)

static constexpr int NB   = 8;      // batch
static constexpr int NTOK = 16384;  // tokens N
static constexpr int NC   = 512;    // channels C
static constexpr int NH   = 8;      // heads
static constexpr int ND   = 64;     // head dim = skip_proj = gf_proj = N1
static constexpr int KCH  = 8;      // token chunks for projection kernel
static constexpr int TOKC = 128;    // token tile for apply kernel

// ---------------------------------------------------------------- zero ws
__global__ void kZero(float* __restrict__ p, int n) {
  int i = blockIdx.x * blockDim.x + threadIdx.x;
  if (i < n) p[i] = 0.0f;
}

// ------------------------------------------------- P1[bh] = skip_headᵀ·EF1
// grid (KCH, 64bh), 128 threads (4 waves). Wave w owns rows i0=16w of the
// 64x64 output, accumulating all 4 k-tiles over its 2048-token chunk.
__global__ void __launch_bounds__(128)
kProj(const float* __restrict__ skip, const float* __restrict__ EF1,
      float* __restrict__ P1) {
  const int chunk = blockIdx.x;           // 0..KCH-1
  const int bh    = blockIdx.y;           // 0..63
  const int b = bh >> 3, h = bh & 7;
  const int tid  = threadIdx.x;
  const int wave = tid >> 5, lane = tid & 31;
  const int llo = lane & 15, lhi = lane >> 4;
  const int i0 = wave * 16;

  // A[M=i, K=token] fragment: lanes 0-15 -> tokens n0,n0+1; lanes 16-31 -> n0+2,n0+3
  const size_t skipBase = (size_t)b * NTOK * NC + (size_t)(h * ND + i0 + llo);

  v8f acc0 = {}, acc1 = {}, acc2 = {}, acc3 = {};
  const int nstart = chunk * (NTOK / KCH);

  for (int s = 0; s < (NTOK / KCH) / 4; ++s) {
    const int n0 = nstart + 4 * s;
    const int ra = n0 + 2 * lhi;

    v2f a;
    a.x = skip[skipBase + (size_t)ra * NC];
    a.y = skip[skipBase + (size_t)(ra + 1) * NC];

    const float* e = EF1 + ra * ND + llo;   // B[K=token, N=k] fragment base
    v2f b0, b1, b2, b3;
    b0.x = e[0];  b0.y = e[ND + 0];
    b1.x = e[16]; b1.y = e[ND + 16];
    b2.x = e[32]; b2.y = e[ND + 32];
    b3.x = e[48]; b3.y = e[ND + 48];

    acc0 = WMMA_F32X4(a, b0, acc0);
    acc1 = WMMA_F32X4(a, b1, acc1);
    acc2 = WMMA_F32X4(a, b2, acc2);
    acc3 = WMMA_F32X4(a, b3, acc3);
  }

  // D layout: VGPR r, lanes 0-15 -> (M=r, N=lane); lanes 16-31 -> (M=r+8, N=lane-16)
  float* base = P1 + (size_t)bh * ND * ND;
#pragma unroll
  for (int r = 0; r < 8; ++r) {
    const int i = i0 + r + 8 * lhi;
    atomicAdd(&base[i * ND + 0  + llo], acc0[r]);
    atomicAdd(&base[i * ND + 16 + llo], acc1[r]);
    atomicAdd(&base[i * ND + 32 + llo], acc2[r]);
    atomicAdd(&base[i * ND + 48 + llo], acc3[r]);
  }
}

// --------------------------- tiny per-(b,h) stage: proj2, l2norm, attn, softmax
// grid 64 blocks, 256 threads: thread = (row i, quarter q); each owns 16 cols.
__global__ void __launch_bounds__(256)
kAttn(const float* __restrict__ gf, const float* __restrict__ EF2,
      const float* __restrict__ P1, const float* __restrict__ temperature,
      float* __restrict__ attn) {
  const int bh = blockIdx.x;
  const int b = bh >> 3, h = bh & 7;
  const int tid = threadIdx.x;
  const int i = tid >> 2, q = tid & 3;

  __shared__ float sGF[64 * 64];   // [n][i]
  __shared__ float sE2[64 * 64];   // [n][k]
  __shared__ float sP1[64 * 65];   // normalized skip proj  [i][t]
  __shared__ float sP2[64 * 65];   // normalized global proj [t][j]
  __shared__ float sRed[64 * 4];

#pragma unroll
  for (int it = 0; it < 16; ++it) {
    int flat = tid + it * 256;             // 4096
    int n = flat >> 6, ii = flat & 63;
    sGF[flat] = gf[((size_t)(b * ND + n)) * NC + h * ND + ii];
    sE2[flat] = EF2[flat];
  }
  __syncthreads();

  // P2 row t=i : global_q_p[t][k] = sum_n gf[n][t] * EF2[n][k]
  float p2[16];
#pragma unroll
  for (int k = 0; k < 16; ++k) p2[k] = 0.0f;
  for (int n = 0; n < 64; ++n) {
    const float g = sGF[n * 64 + i];
#pragma unroll
    for (int k = 0; k < 16; ++k) p2[k] += g * sE2[n * 64 + q * 16 + k];
  }
  float ss = 0.0f;
#pragma unroll
  for (int k = 0; k < 16; ++k) ss += p2[k] * p2[k];
  sRed[i * 4 + q] = ss;
  __syncthreads();
  float inv = 1.0f / fmaxf(sqrtf(sRed[i*4] + sRed[i*4+1] + sRed[i*4+2] + sRed[i*4+3]), 1e-12f);
#pragma unroll
  for (int k = 0; k < 16; ++k) sP2[i * 65 + q * 16 + k] = p2[k] * inv;

  // P1 row i (from workspace), normalize over t
  float p1[16];
#pragma unroll
  for (int k = 0; k < 16; ++k) p1[k] = P1[((size_t)bh * ND + i) * ND + q * 16 + k];
  ss = 0.0f;
#pragma unroll
  for (int k = 0; k < 16; ++k) ss += p1[k] * p1[k];
  __syncthreads();                 // sRed reads above done
  sRed[i * 4 + q] = ss;
  __syncthreads();
  inv = 1.0f / fmaxf(sqrtf(sRed[i*4] + sRed[i*4+1] + sRed[i*4+2] + sRed[i*4+3]), 1e-12f);
#pragma unroll
  for (int k = 0; k < 16; ++k) sP1[i * 65 + q * 16 + k] = p1[k] * inv;
  __syncthreads();

  // attn[i][j] = sum_t sP1[i][t] * sP2[t][j], * temperature[h]
  const float T = temperature[h];
  float a[16];
#pragma unroll
  for (int j = 0; j < 16; ++j) a[j] = 0.0f;
  for (int t = 0; t < 64; ++t) {
    const float x = sP1[i * 65 + t];
#pragma unroll
    for (int j = 0; j < 16; ++j) a[j] += x * sP2[t * 65 + q * 16 + j];
  }
#pragma unroll
  for (int j = 0; j < 16; ++j) a[j] *= T;

  // softmax along j (row i)
  float m = -3.0e38f;
#pragma unroll
  for (int j = 0; j < 16; ++j) m = fmaxf(m, a[j]);
  sRed[i * 4 + q] = m;
  __syncthreads();
  m = fmaxf(fmaxf(sRed[i*4], sRed[i*4+1]), fmaxf(sRed[i*4+2], sRed[i*4+3]));
  float s = 0.0f;
#pragma unroll
  for (int j = 0; j < 16; ++j) { a[j] = expf(a[j] - m); s += a[j]; }
  __syncthreads();
  sRed[i * 4 + q] = s;
  __syncthreads();
  s = sRed[i*4] + sRed[i*4+1] + sRed[i*4+2] + sRed[i*4+3];
  const float rs = 1.0f / s;
#pragma unroll
  for (int j = 0; j < 16; ++j)
    attn[(size_t)bh * ND * ND + i * ND + q * 16 + j] = a[j] * rs;
}

// -------------------------------------------- y[b,n,h,:] = attn[bh] @ out[b,n,h,:]
// grid (128 chunks, 64bh), 256 threads (8 waves). LDS-stage 128 tokens x 64ch
// (stride 65 -> conflict-free), tokens as WMMA M dim so stores are contiguous.
__global__ void __launch_bounds__(256)
kApply(const float* __restrict__ outv, const float* __restrict__ attn,
       float* __restrict__ y) {
  const int chunk = blockIdx.x;          // 0..127
  const int bh    = blockIdx.y;          // 0..63
  const int b = bh >> 3, h = bh & 7;
  const int tid = threadIdx.x;

  __shared__ float sT[TOKC * 65];

  const size_t gbase = ((size_t)b * NTOK + (size_t)chunk * TOKC) * NC + h * ND;
#pragma unroll
  for (int it = 0; it < 8; ++it) {
    int flat = tid + it * 256;           // 2048 float4 tiles
    int row = flat >> 4, f4 = flat & 15;
    const float4 v = *(const float4*)(outv + gbase + (size_t)row * NC + f4 * 4);
    float* d = &sT[row * 65 + f4 * 4];
    d[0] = v.x; d[1] = v.y; d[2] = v.z; d[3] = v.w;
  }

  const int wave = tid >> 5, lane = tid & 31;
  const int llo = lane & 15, lhi = lane >> 4;
  const int i0   = (wave & 3) * 16;      // output-channel tile
  const int tsub = (wave >> 2) * 64;     // token sub-block

  // Preload B fragments: B[K=j, N=i] = attn[i][j]  (tiny, L2-hot)
  v2f bf[16];
  const float* ab = attn + (size_t)bh * ND * ND;
#pragma unroll
  for (int s = 0; s < 16; ++s) {
    const int row = i0 + llo, col = 4 * s + 2 * lhi;
    bf[s].x = ab[row * ND + col];
    bf[s].y = ab[row * ND + col + 1];
  }
  __syncthreads();

#pragma unroll
  for (int tt = 0; tt < 4; ++tt) {
    const int t0 = tsub + tt * 16;
    v8f acc = {};
    const float* arow = &sT[(t0 + llo) * 65];   // A[M=token, K=j]
#pragma unroll
    for (int s = 0; s < 16; ++s) {
      const int col = 4 * s + 2 * lhi;
      v2f a;
      a.x = arow[col];
      a.y = arow[col + 1];
      acc = WMMA_F32X4(a, bf[s], acc);
    }
    // D[M=token, N=i]: VGPR r -> tokens t0+r (lanes 0-15) / t0+r+8 (lanes 16-31)
    const size_t ob = ((size_t)b * NTOK + (size_t)chunk * TOKC + t0 + 8 * lhi) * NC
                      + h * ND + i0 + llo;
#pragma unroll
    for (int r = 0; r < 8; ++r) y[ob + (size_t)r * NC] = acc[r];
  }
}

extern "C" void kernel_launch(void* const* d_in, const int* in_sizes, int n_in,
                              void* d_out, int out_size, void* d_ws, size_t ws_size,
                              hipStream_t stream) {
  const float* skip = (const float*)d_in[0];
  const float* outv = (const float*)d_in[1];
  const float* gf   = (const float*)d_in[2];
  const float* EF1  = (const float*)d_in[3];
  const float* EF2  = (const float*)d_in[4];
  const float* temp = (const float*)d_in[5];
  float* y = (float*)d_out;

  float* P1 = (float*)d_ws;                 // 64 * 64*64 partial sums (1 MB)
  float* AT = P1 + 64 * ND * ND;            // 64 * 64*64 softmaxed attn (1 MB)

  const int np1 = 64 * ND * ND;
  kZero<<<(np1 + 255) / 256, 256, 0, stream>>>(P1, np1);
  kProj<<<dim3(KCH, 64), 128, 0, stream>>>(skip, EF1, P1);
  kAttn<<<64, 256, 0, stream>>>(gf, EF2, P1, temp, AT);
  kApply<<<dim3(NTOK / TOKC, 64), 256, 0, stream>>>(outv, AT, y);
}